// ProbabilityDropout_2293512536898
// MI455X (gfx1250) — compile-verified
//
#include <hip/hip_runtime.h>
#include <math.h>
#include <stdint.h>

#define NCOLS 4096
#define NBINS 4096
#define NROWS 2048
#define ZERO_POINT 0.0003f

// ---------------- Kernel 1: partial column sums / sums of squares (float4-wide) ----------------
constexpr int K1_BLOCK = 256;   // threads; each handles 4 columns -> 1024 columns per block

__global__ __launch_bounds__(K1_BLOCK)
void colstats_partial(const float* __restrict__ z,
                      float* __restrict__ psum,
                      float* __restrict__ psq,
                      int rowsPerChunk) {
  const int c4 = blockIdx.x * K1_BLOCK + threadIdx.x;   // float4 column index
  const int r0 = blockIdx.y * rowsPerChunk;
  const float4* p = reinterpret_cast<const float4*>(z) + (size_t)r0 * (NCOLS / 4) + c4;
  float4 s = make_float4(0.f, 0.f, 0.f, 0.f);
  float4 q = make_float4(0.f, 0.f, 0.f, 0.f);
  #pragma unroll 4
  for (int r = 0; r < rowsPerChunk; ++r) {
    float4 v = p[(size_t)r * (NCOLS / 4)];
    s.x += v.x; s.y += v.y; s.z += v.z; s.w += v.w;
    q.x = fmaf(v.x, v.x, q.x); q.y = fmaf(v.y, v.y, q.y);
    q.z = fmaf(v.z, v.z, q.z); q.w = fmaf(v.w, v.w, q.w);
  }
  const size_t o = (size_t)blockIdx.y * (NCOLS / 4) + c4;
  reinterpret_cast<float4*>(psum)[o] = s;
  reinterpret_cast<float4*>(psq)[o]  = q;
}

// ---------------- Kernel 2: finalize mu and scale = exp(0.5*var) ----------------
__global__ __launch_bounds__(256)
void colstats_final(const float* __restrict__ psum,
                    const float* __restrict__ psq,
                    float* __restrict__ mu,
                    float* __restrict__ scale,
                    int rch) {
  const int c4 = blockIdx.x * blockDim.x + threadIdx.x;  // float4 column index
  float4 s = make_float4(0.f, 0.f, 0.f, 0.f);
  float4 q = make_float4(0.f, 0.f, 0.f, 0.f);
  for (int c = 0; c < rch; ++c) {
    float4 a = reinterpret_cast<const float4*>(psum)[(size_t)c * (NCOLS / 4) + c4];
    float4 b = reinterpret_cast<const float4*>(psq)[(size_t)c * (NCOLS / 4) + c4];
    s.x += a.x; s.y += a.y; s.z += a.z; s.w += a.w;
    q.x += b.x; q.y += b.y; q.z += b.z; q.w += b.w;
  }
  const float inv = 1.0f / NROWS;
  float4 m, sc;
  m.x = s.x * inv; m.y = s.y * inv; m.z = s.z * inv; m.w = s.w * inv;
  sc.x = expf(0.5f * fmaxf(q.x * inv - m.x * m.x, 0.0f));
  sc.y = expf(0.5f * fmaxf(q.y * inv - m.y * m.y, 0.0f));
  sc.z = expf(0.5f * fmaxf(q.z * inv - m.z * m.z, 0.0f));
  sc.w = expf(0.5f * fmaxf(q.w * inv - m.w * m.w, 0.0f));
  reinterpret_cast<float4*>(mu)[c4]    = m;
  reinterpret_cast<float4*>(scale)[c4] = sc;
}

// ---------------- Kernel 3: per-row zs -> histogram -> softmax -> threshold -> x* ----------------
constexpr int K3_T = 1024;   // 32 wave32 waves; each thread owns columns 4t..4t+3

__global__ __launch_bounds__(K3_T)
void row_hist_softmax(const float* __restrict__ eps,
                      const float* __restrict__ x,
                      const float* __restrict__ mu,
                      const float* __restrict__ scale,
                      float* __restrict__ out) {
  __shared__ __align__(16) unsigned s_hist[NBINS];   // 16 KB
  __shared__ __align__(16) float s_x[NCOLS];         // 16 KB, filled by async DMA
  __shared__ float s_r0[32];
  __shared__ float s_r1[32];
  __shared__ float s_bcast[2];

  const int t    = threadIdx.x;
  const int wave = t >> 5;
  const int lane = t & 31;
  const size_t base = (size_t)blockIdx.x * NCOLS;

  // ---- issue async global->LDS copy of this row of x (ASYNCcnt); each thread copies
  //      exactly the 16B it will read back at the end. Fetch hides behind all work below.
  {
    unsigned lds_off = (unsigned)(uintptr_t)(const void*)(&s_x[t * 4]);
    const float* gp  = x + base + (size_t)t * 4;
    asm volatile("global_load_async_to_lds_b128 %0, %1, off"
                 :: "v"(lds_off), "v"(gp) : "memory");
  }

  // ---- phase 1: zs = mu + scale*eps (b128 loads, values kept in registers), zero hist
  const float4 e4  = reinterpret_cast<const float4*>(eps + base)[t];
  const float4 m4  = reinterpret_cast<const float4*>(mu)[t];
  const float4 sc4 = reinterpret_cast<const float4*>(scale)[t];
  float zv[4];
  zv[0] = fmaf(sc4.x, e4.x, m4.x);
  zv[1] = fmaf(sc4.y, e4.y, m4.y);
  zv[2] = fmaf(sc4.z, e4.z, m4.z);
  zv[3] = fmaf(sc4.w, e4.w, m4.w);
  reinterpret_cast<uint4*>(s_hist)[t] = make_uint4(0u, 0u, 0u, 0u);  // ds_store_b128

  float vmin = fminf(fminf(zv[0], zv[1]), fminf(zv[2], zv[3]));
  float vmax = fmaxf(fmaxf(zv[0], zv[1]), fmaxf(zv[2], zv[3]));

  // ---- block min/max reduction (wave32 shuffle tree + one LDS hop)
  #pragma unroll
  for (int o = 16; o > 0; o >>= 1) {
    vmin = fminf(vmin, __shfl_xor(vmin, o, 32));
    vmax = fmaxf(vmax, __shfl_xor(vmax, o, 32));
  }
  if (lane == 0) { s_r0[wave] = vmin; s_r1[wave] = vmax; }
  __syncthreads();
  if (wave == 0) {
    float m = s_r0[lane], M = s_r1[lane];
    #pragma unroll
    for (int o = 16; o > 0; o >>= 1) {
      m = fminf(m, __shfl_xor(m, o, 32));
      M = fmaxf(M, __shfl_xor(M, o, 32));
    }
    if (lane == 0) { s_bcast[0] = m; s_bcast[1] = M; }
  }
  __syncthreads();
  const float rmin = s_bcast[0];
  float width = s_bcast[1] - rmin;
  if (!(width > 0.0f)) width = 1.0f;   // degenerate-range guard (matches reference)

  // ---- phase 2: histogram via LDS atomics (ds_add_u32)
  #pragma unroll
  for (int i = 0; i < 4; ++i) {
    int idx = (int)floorf((float)NBINS * (zv[i] - rmin) / width);
    idx = idx < 0 ? 0 : (idx > NBINS - 1 ? NBINS - 1 : idx);
    atomicAdd(&s_hist[idx], 1u);
  }
  __syncthreads();

  // ---- phase 3: softmax over counts (max-subtracted); counts read back as b128
  const uint4 c4v = reinterpret_cast<const uint4*>(s_hist)[t];
  unsigned cv[4] = { c4v.x, c4v.y, c4v.z, c4v.w };
  unsigned cmax = cv[0];
  cmax = cv[1] > cmax ? cv[1] : cmax;
  cmax = cv[2] > cmax ? cv[2] : cmax;
  cmax = cv[3] > cmax ? cv[3] : cmax;
  #pragma unroll
  for (int o = 16; o > 0; o >>= 1) {
    unsigned oth = __shfl_xor(cmax, o, 32);
    cmax = oth > cmax ? oth : cmax;
  }
  if (lane == 0) s_r0[wave] = (float)cmax;
  __syncthreads();
  if (wave == 0) {
    float M = s_r0[lane];
    #pragma unroll
    for (int o = 16; o > 0; o >>= 1) M = fmaxf(M, __shfl_xor(M, o, 32));
    if (lane == 0) s_bcast[0] = M;
  }
  __syncthreads();
  const float cmaxf = s_bcast[0];

  float ev[4];
  float lsum = 0.0f;
  #pragma unroll
  for (int i = 0; i < 4; ++i) {
    ev[i] = expf((float)cv[i] - cmaxf);
    lsum += ev[i];
  }
  #pragma unroll
  for (int o = 16; o > 0; o >>= 1) lsum += __shfl_xor(lsum, o, 32);
  if (lane == 0) s_r1[wave] = lsum;
  __syncthreads();
  if (wave == 0) {
    float S = s_r1[lane];
    #pragma unroll
    for (int o = 16; o > 0; o >>= 1) S += __shfl_xor(S, o, 32);
    if (lane == 0) s_bcast[1] = S;
  }

  // ---- drain the async x-row copy; barrier also publishes block-wide sum
  asm volatile("s_wait_asynccnt 0" ::: "memory");
  __syncthreads();
  const float sum = s_bcast[1];

  const float4 xr = reinterpret_cast<const float4*>(s_x)[t];  // ds_load_b128
  float4 o4;
  float p;
  p = ev[0] / sum; if (p < ZERO_POINT) p = 0.0f; o4.x = xr.x * p;
  p = ev[1] / sum; if (p < ZERO_POINT) p = 0.0f; o4.y = xr.y * p;
  p = ev[2] / sum; if (p < ZERO_POINT) p = 0.0f; o4.z = xr.z * p;
  p = ev[3] / sum; if (p < ZERO_POINT) p = 0.0f; o4.w = xr.w * p;
  reinterpret_cast<float4*>(out + base)[t] = o4;              // global_store_b128
}

// ---------------- launcher ----------------
extern "C" void kernel_launch(void* const* d_in, const int* in_sizes, int n_in,
                              void* d_out, int out_size, void* d_ws, size_t ws_size,
                              hipStream_t stream) {
  const float* z   = (const float*)d_in[0];
  const float* x   = (const float*)d_in[1];
  const float* eps = (const float*)d_in[2];
  float* out = (float*)d_out;

  // Pick row-chunk count for the stats pass from the available workspace
  // (needs rch*NCOLS*2 + 2*NCOLS floats). Powers of two divide NROWS evenly.
  int rch = 64;
  while (rch > 1 &&
         ((size_t)rch * NCOLS * 2 + 2 * NCOLS) * sizeof(float) > ws_size) {
    rch >>= 1;
  }
  const int rowsPerChunk = NROWS / rch;

  float* ws    = (float*)d_ws;
  float* psum  = ws;                                  // rch * NCOLS
  float* psq   = psum + (size_t)rch * NCOLS;          // rch * NCOLS
  float* mu    = psq  + (size_t)rch * NCOLS;          // NCOLS
  float* scale = mu + NCOLS;                          // NCOLS

  colstats_partial<<<dim3(NCOLS / 4 / K1_BLOCK, rch), K1_BLOCK, 0, stream>>>(
      z, psum, psq, rowsPerChunk);
  colstats_final<<<dim3(NCOLS / 4 / 256), 256, 0, stream>>>(psum, psq, mu, scale, rch);
  row_hist_softmax<<<dim3(NROWS), K3_T, 0, stream>>>(eps, x, mu, scale, out);
}